// VectorQuantizer_41901700940501
// MI455X (gfx1250) — compile-verified
//
#include <hip/hip_runtime.h>

typedef float v2f __attribute__((ext_vector_type(2)));
typedef float v8f __attribute__((ext_vector_type(8)));
typedef unsigned int u32x4 __attribute__((ext_vector_type(4)));
typedef int i32x4 __attribute__((ext_vector_type(4)));
typedef int i32x8 __attribute__((ext_vector_type(8)));

#define K_CODES 1024
#define DIM     64
#define CHUNK   512          // codes resident in LDS per pass
#define PADC    68           // 64 channels + ||e||^2 + 3 pad (conflict-free DS, b64-aligned)
#define NPOS    131072       // 32*64*64 positions
#define QELEMS  8388608      // 32*64*64*64

// ---------------- kernel 0: code norms ----------------
__global__ void vq_norms(const float* __restrict__ emb, float* __restrict__ norms) {
    int code = blockIdx.x * blockDim.x + threadIdx.x;
    if (code < K_CODES) {
        const float4* row = (const float4*)(emb + (size_t)code * DIM);
        float s = 0.f;
        #pragma unroll
        for (int i = 0; i < 16; ++i) {
            float4 v = row[i];
            s += v.x * v.x + v.y * v.y + v.z * v.z + v.w * v.w;
        }
        norms[code] = s;
    }
}

// ---------------- kernel 1: WMMA argmin with TDM-staged codebook ----------------
// score(n,k) = ||e_k||^2 - 2 x_n . e_k   via a 17-step K=4 fp32 WMMA chain (norm folded as channel 64)
__launch_bounds__(256, 1)
__global__ void vq_argmin(const float* __restrict__ in, const float* __restrict__ emb,
                          const float* __restrict__ norms, int* __restrict__ idxOut) {
    __shared__ alignas(128) float sB[CHUNK * PADC];   // 139264 bytes, row stride 68 floats

    const int tid    = threadIdx.x;
    const int lane   = tid & 31;
    const int wave   = tid >> 5;
    const int laneLo = lane & 15;
    const bool hi    = lane >= 16;
    const int n0     = blockIdx.x * 256 + wave * 32;   // 32 positions per wave (2 M-tiles)

    // ---- A operand (positions) in registers: -2x, plus norm-pickup channel ----
    v2f A[2][17];
    #pragma unroll
    for (int at = 0; at < 2; ++at) {
        int n  = n0 + at * 16 + laneLo;
        int b  = n >> 12;
        int hw = n & 4095;
        const float* base = in + ((size_t)b << 18) + hw;
        #pragma unroll
        for (int s = 0; s < 16; ++s) {
            int c0 = 4 * s + (hi ? 2 : 0);
            A[at][s].x = -2.0f * base[(size_t)c0 << 12];
            A[at][s].y = -2.0f * base[(size_t)(c0 + 1) << 12];
        }
        A[at][16].x = hi ? 0.0f : 1.0f;   // channels 64..67: picks up ||e||^2 row
        A[at][16].y = 0.0f;
    }

    float    minv[2][8];
    unsigned mini[2][8];
    #pragma unroll
    for (int at = 0; at < 2; ++at)
        #pragma unroll
        for (int r = 0; r < 8; ++r) { minv[at][r] = 3.4e38f; mini[at][r] = 0u; }

    // per-wave uniform slice for the TDM copy: 64 codebook rows per wave
    const unsigned wslice   = (unsigned)__builtin_amdgcn_readfirstlane(wave);
    const unsigned ldsBase  = (unsigned)(uintptr_t)(void*)&sB[0];

    for (int cb = 0; cb < K_CODES; cb += CHUNK) {
        __syncthreads();   // previous chunk fully consumed before overwrite

        // ---- Tensor Data Mover: DMA 64 rows x 64 dwords into LDS with HW padding ----
        // pad_enable, pad_interval=5 (insert after every 64 DWORDs), pad_amount=3 (4 DWORDs)
        // -> destination row stride 68 floats, matching PADC.
        {
            unsigned long long gaddr =
                (unsigned long long)(uintptr_t)(emb + (size_t)(cb + wslice * 64) * DIM);
            unsigned laddr = ldsBase + wslice * 64u * PADC * 4u;

            u32x4 g0;
            g0[0] = 1u;                                   // count=1 (valid descriptor)
            g0[1] = laddr;                                // lds_addr
            g0[2] = (unsigned)gaddr;                      // global_addr[31:0]
            g0[3] = (unsigned)((gaddr >> 32) & 0x1FFFFFFu) | (2u << 30);  // addr[56:32] | type=2

            i32x8 g1;
            g1[0] = (2 << 16) | (1 << 20) | (5 << 22) | (3 << 25); // data_size=4B, pad_en, interval, amount
            g1[1] = (int)(64u << 16);                     // tensor_dim0 = 64 (low 16 in [31:16])
            g1[2] = (int)(1024u << 16);                   // tensor_dim0 hi=0 | tensor_dim1 = 1024
            g1[3] = (int)(64u << 16);                     // tensor_dim1 hi=0 | tile_dim0 = 64
            g1[4] = 64;                                   // tile_dim1 = 64 rows | tile_dim2 = 0
            g1[5] = 64;                                   // tensor_dim0_stride = 64 (row stride)
            g1[6] = 0;
            g1[7] = 0;

            i32x4 g2 = {0, 0, 0, 0};
            i32x4 g3 = {0, 0, 0, 0};
            i32x8 g4 = {0, 0, 0, 0, 0, 0, 0, 0};
            __builtin_amdgcn_tensor_load_to_lds(g0, g1, g2, g3, g4, 0);
        }

        // stage norm column + zero pad with plain DS stores (disjoint from TDM region)
        for (int i = tid; i < CHUNK; i += 256) {
            sB[i * PADC + 64] = norms[cb + i];
            sB[i * PADC + 65] = 0.f;
            sB[i * PADC + 66] = 0.f;
            sB[i * PADC + 67] = 0.f;
        }

        __builtin_amdgcn_s_wait_tensorcnt(0);   // this wave's DMA complete
        __syncthreads();                        // publish all waves' slices + norms

        for (int t = 0; t < CHUNK / 16; ++t) {
            v2f B[17];
            const float* bp = &sB[(t * 16 + laneLo) * PADC + (hi ? 2 : 0)];
            #pragma unroll
            for (int s = 0; s < 17; ++s) {
                B[s].x = bp[4 * s];
                B[s].y = bp[4 * s + 1];
            }

            v8f D0 = {0.f,0.f,0.f,0.f,0.f,0.f,0.f,0.f};
            v8f D1 = {0.f,0.f,0.f,0.f,0.f,0.f,0.f,0.f};
            #pragma unroll
            for (int s = 0; s < 17; ++s) {
                D0 = __builtin_amdgcn_wmma_f32_16x16x4_f32(false, A[0][s], false, B[s],
                                                           (short)0, D0, false, false);
                D1 = __builtin_amdgcn_wmma_f32_16x16x4_f32(false, A[1][s], false, B[s],
                                                           (short)0, D1, false, false);
            }

            unsigned codeBase = (unsigned)(cb + t * 16 + laneLo);
            #pragma unroll
            for (int r = 0; r < 8; ++r) {
                float s0 = D0[r], s1 = D1[r];
                if (s0 < minv[0][r]) { minv[0][r] = s0; mini[0][r] = codeBase; }
                if (s1 < minv[1][r]) { minv[1][r] = s1; mini[1][r] = codeBase; }
            }
        }
    }

    // reduce across the 16 lanes holding the same M row (xor masks stay within half-wave)
    #pragma unroll
    for (int at = 0; at < 2; ++at)
        #pragma unroll
        for (int r = 0; r < 8; ++r) {
            float v = minv[at][r];
            int   ix = (int)mini[at][r];
            #pragma unroll
            for (int m = 1; m < 16; m <<= 1) {
                float ov = __shfl_xor(v, m, 32);
                int   oi = __shfl_xor(ix, m, 32);
                if (ov < v || (ov == v && oi < ix)) { v = ov; ix = oi; }
            }
            mini[at][r] = (unsigned)ix;
        }

    if (laneLo == 0) {   // lanes 0..15 hold M=r, lanes 16..31 hold M=8+r
        #pragma unroll
        for (int at = 0; at < 2; ++at)
            #pragma unroll
            for (int r = 0; r < 8; ++r)
                idxOut[n0 + at * 16 + (hi ? 8 : 0) + r] = (int)mini[at][r];
    }
}

// ---------------- kernel 2: quantized output (b,c,h,w), perfectly coalesced stores ----------------
__global__ void vq_quantize_out(const float* __restrict__ emb, const int* __restrict__ idx,
                                float* __restrict__ qout) {
    int t  = blockIdx.x * blockDim.x + threadIdx.x;   // < 8388608
    int b  = t >> 18;
    int c  = (t >> 12) & 63;
    int hw = t & 4095;
    int n  = (b << 12) | hw;
    int i  = idx[n];
    qout[t] = emb[i * DIM + c];
}

// ---------------- kernel 3: loss (b,h,w,c) via LDS transpose + idx as float ----------------
__launch_bounds__(256, 1)
__global__ void vq_loss(const float* __restrict__ in, const float* __restrict__ emb,
                        const int* __restrict__ idx, float* __restrict__ lossOut,
                        float* __restrict__ idxfOut) {
    __shared__ float sx[256 * 65];
    const int tid = threadIdx.x;
    const int n0  = blockIdx.x * 256;               // block covers 256 positions, same b
    const int b   = n0 >> 12;
    const int hw0 = n0 & 4095;

    const float* base = in + ((size_t)b << 18) + hw0;
    for (int c = 0; c < 64; ++c)                    // coalesced reads, own-row LDS writes
        sx[tid * 65 + c] = base[((size_t)c << 12) + tid];

    const int n = n0 + tid;
    const int i = idx[n];
    idxfOut[n] = (float)i;

    const float4* er = (const float4*)(emb + (size_t)i * DIM);
    #pragma unroll
    for (int c4 = 0; c4 < 16; ++c4) {
        float4 e = er[c4];
        float* xr = &sx[tid * 65 + c4 * 4];
        float d0 = e.x - xr[0], d1 = e.y - xr[1], d2 = e.z - xr[2], d3 = e.w - xr[3];
        xr[0] = 1.25f * d0 * d0;
        xr[1] = 1.25f * d1 * d1;
        xr[2] = 1.25f * d2 * d2;
        xr[3] = 1.25f * d3 * d3;
    }
    __syncthreads();

    float* lbase = lossOut + (size_t)n0 * 64;       // contiguous 64KB region for this block
    for (int k = 0; k < 64; ++k) {                  // coalesced stores, conflict-free LDS reads
        int j = k * 256 + tid;
        lbase[j] = sx[(j >> 6) * 65 + (j & 63)];
    }
}

extern "C" void kernel_launch(void* const* d_in, const int* in_sizes, int n_in,
                              void* d_out, int out_size, void* d_ws, size_t ws_size,
                              hipStream_t stream) {
    const float* in  = (const float*)d_in[0];   // (32,64,64,64) b c h w
    const float* emb = (const float*)d_in[1];   // (1024,64)

    float* norms = (float*)d_ws;                          // 1024 floats
    int*   idxws = (int*)((char*)d_ws + 4096);            // 131072 ints

    float* qout = (float*)d_out;                          // [8388608] (b,c,h,w)
    float* loss = qout + QELEMS;                          // [8388608] (b,h,w,c)
    float* idxf = qout + 2 * QELEMS;                      // [131072]

    vq_norms<<<K_CODES / 256, 256, 0, stream>>>(emb, norms);
    vq_argmin<<<NPOS / 256, 256, 0, stream>>>(in, emb, norms, idxws);
    vq_quantize_out<<<QELEMS / 256, 256, 0, stream>>>(emb, idxws, qout);
    vq_loss<<<NPOS / 256, 256, 0, stream>>>(in, emb, idxws, loss, idxf);
}